// RWSEEdgeEncoder_17377437679647
// MI455X (gfx1250) — compile-verified
//
#include <hip/hip_runtime.h>
#include <hip/hip_bf16.h>

typedef __attribute__((ext_vector_type(2))) float v2f;
typedef __attribute__((ext_vector_type(8))) float v8f;

// Fused gather + (16 -> 64) GEMM + bias, using V_WMMA_F32_16X16X4_F32.
// One wave32 processes a tile of 16 edges:
//   A (16x4 f32)  : gathered RWSE rows, lane l holds row m=l%16, k=4s+2*(l/16)+{0,1}
//   B (4x16 f32)  : W[k][n], lane l holds n=16t+l%16, same k striping
//   C/D (16x16)   : vgpr v, lane l  <->  (row v+8*(l/16), col l%16)
// SYM=true averages the (r,c) and (c,r) rows before the matmul (linearity of x@W+b).
template <bool SYM>
__global__ __launch_bounds__(256) void rwse_edge_gemm(
    const float* __restrict__ X,     // (B*Nd*Nd, 16) row-major, Nd = 1<<ldShift
    const int*   __restrict__ idx,   // (2, nEdges) flat: src row then dst row
    const float* __restrict__ W,     // (16, 64) row-major
    const float* __restrict__ bias,  // (64,)
    float*       __restrict__ out,   // (nEdges, 64)
    int nEdges, int ldShift)
{
    const int Nd   = 1 << ldShift;
    const int mask = Nd - 1;
    const int lane = threadIdx.x & 31;
    const int wave = threadIdx.x >> 5;
    const int tile = blockIdx.x * (blockDim.x >> 5) + wave;  // 16 edges per wave
    const int m    = lane & 15;   // M index (edge within tile) / N index (col within tile)
    const int half = lane >> 4;   // which half of the wave

    // ---- per-edge index math (lanes 0..15 and 16..31 mirror the same edge) ----
    const int e   = tile * 16 + m;
    const int src = idx[e];
    const int dst = idx[nEdges + e];
    const int g = src >> ldShift;
    const int r = src & mask;
    const int c = dst & mask;

    const float* __restrict__ p0 = X + (((g * Nd + r) * Nd + c) << 4);
    const float* __restrict__ p1 = SYM ? (X + (((g * Nd + c) * Nd + r) << 4)) : nullptr;

    // ---- B fragments: lane holds W[k=4s+2*half+v][n=16t+m], v in {0,1} ----
    v2f bfrag[4][4];
#pragma unroll
    for (int t = 0; t < 4; ++t) {
#pragma unroll
        for (int s = 0; s < 4; ++s) {
            const int k = 4 * s + 2 * half;
            const int n = 16 * t + m;
            v2f b;
            b[0] = W[k * 64 + n];
            b[1] = W[(k + 1) * 64 + n];
            bfrag[t][s] = b;
        }
    }

    // ---- A fragments: one aligned float2 per K-step (row base is 64B aligned, k even) ----
    v2f afrag[4];
#pragma unroll
    for (int s = 0; s < 4; ++s) {
        const int k = 4 * s + 2 * half;
        v2f a = *(const v2f*)(p0 + k);
        if (SYM) {
            v2f a2 = *(const v2f*)(p1 + k);
            a = (a + a2) * 0.5f;
        }
        afrag[s] = a;
    }

    // ---- accumulators start at the bias (column-broadcast across all 8 row-VGPRs) ----
    v8f acc[4];
#pragma unroll
    for (int t = 0; t < 4; ++t) {
        const float bv = bias[16 * t + m];
#pragma unroll
        for (int v = 0; v < 8; ++v) acc[t][v] = bv;
    }

    // ---- 4 N-tiles x 4 K-steps of fp32 WMMA ----
#pragma unroll
    for (int t = 0; t < 4; ++t) {
#pragma unroll
        for (int s = 0; s < 4; ++s) {
            acc[t] = __builtin_amdgcn_wmma_f32_16x16x4_f32(
                /*neg_a=*/false, afrag[s],
                /*neg_b=*/false, bfrag[t][s],
                /*c_mod=*/(short)0, acc[t],
                /*reuse_a=*/false, /*reuse_b=*/false);
        }
    }

    // ---- store: out[(tile*16 + v + 8*half)*64 + 16t + m] ----
    float* ob = out + ((tile * 16 + 8 * half) * 64 + m);
#pragma unroll
    for (int t = 0; t < 4; ++t) {
#pragma unroll
        for (int v = 0; v < 8; ++v) {
            ob[v * 64 + 16 * t] = acc[t][v];
        }
    }
}

extern "C" void kernel_launch(void* const* d_in, const int* in_sizes, int n_in,
                              void* d_out, int out_size, void* d_ws, size_t ws_size,
                              hipStream_t stream) {
    (void)in_sizes; (void)n_in; (void)d_ws; (void)ws_size; (void)out_size;

    constexpr int B = 16, N = 128, E = 512, M = 4096, D = 64;

    const float* edge_RWSE     = (const float*)d_in[0];  // (B*N*N, 16)
    const float* e2e_edge_RWSE = (const float*)d_in[1];  // (B*E*E, 16)
    // d_in[2] = batch, d_in[3] = e_batch : implied by equal-sized graphs, unused
    const int*   edge_index     = (const int*)d_in[4];   // (2, B*E)
    const int*   e2e_edge_index = (const int*)d_in[5];   // (2, B*M)
    const float* W1 = (const float*)d_in[6];             // (16, 64)
    const float* b1 = (const float*)d_in[7];             // (64,)
    const float* W2 = (const float*)d_in[8];             // (16, 64)
    const float* b2 = (const float*)d_in[9];             // (64,)

    float* out1 = (float*)d_out;                 // edge_attr      (B*E, 64)
    float* out2 = out1 + (size_t)B * E * D;      // e2e_edge_attr  (B*M, 64)

    const int nE1 = B * E;   // 8192  edges -> 512 wave-tiles -> 64 blocks of 8 waves
    const int nE2 = B * M;   // 65536 edges -> 4096 wave-tiles -> 512 blocks

    rwse_edge_gemm<true ><<<(nE1 / 16) / 8, 256, 0, stream>>>(
        edge_RWSE, edge_index, W1, b1, out1, nE1, /*log2(N)=*/7);
    rwse_edge_gemm<false><<<(nE2 / 16) / 8, 256, 0, stream>>>(
        e2e_edge_RWSE, e2e_edge_index, W2, b2, out2, nE2, /*log2(E)=*/9);
}